// PlayerLevelGATv2_74929999446104
// MI455X (gfx1250) — compile-verified
//
#include <hip/hip_runtime.h>

// PlayerLevelGATv2 on MI455X (gfx1250, wave32).
// Memory/atomics-bound edge pipeline in fp32; dense layer-2 node transform
// uses V_WMMA_F32_16X16X4_F32 (full-precision fp32 matrix pipe, K=128 = 32 steps).

#define NN     50000
#define EE     800000
#define IN_F   10
#define ED_F   4
#define HIDC   32
#define NHEADS 4
#define F1     128   // NHEADS*HIDC

typedef __attribute__((ext_vector_type(2))) float v2f;
typedef __attribute__((ext_vector_type(8))) float v8f;

__device__ __forceinline__ float warp_sum(float v) {
#pragma unroll
  for (int off = 16; off > 0; off >>= 1) v += __shfl_xor(v, off, 32);
  return v;
}

__device__ __forceinline__ void atomicMaxF(float* addr, float val) {
  int* ai = reinterpret_cast<int*>(addr);
  int old = __float_as_int(*addr);
  while (__int_as_float(old) < val) {
    int assumed = old;
    old = atomicCAS(ai, assumed, __float_as_int(val));
    if (old == assumed) break;
  }
}

// ---------------------------------------------------------------- init
__global__ void init1_kernel(float* acc1, float* amax1, float* asum1) {
  int i = blockIdx.x * 256 + threadIdx.x;          // grid covers NN*F1
  if (i < NN * F1) acc1[i] = 0.f;
  if (i < NN * NHEADS) { amax1[i] = -__builtin_inff(); asum1[i] = 0.f; }
}

__global__ void init2_kernel(float* acc2, float* amax2, float* asum2,
                             float* gsum, float* gmax) {
  int i = blockIdx.x * 256 + threadIdx.x;          // grid covers NN*HIDC
  if (i < NN * HIDC) acc2[i] = 0.f;
  if (i < NN) { amax2[i] = -__builtin_inff(); asum2[i] = 0.f; }
  if (i < HIDC) { gsum[i] = 0.f; gmax[i] = -__builtin_inff(); }
}

// ------------------------------------------------- layer-1 node transform
// xl1 = x@Wl1+bl1 ; xr1 = x@Wr1+br1   (K=10: scalar FMA, weights in LDS)
__global__ void gemm1_kernel(const float* __restrict__ x,
                             const float* __restrict__ Wl, const float* __restrict__ bl,
                             const float* __restrict__ Wr, const float* __restrict__ br,
                             float* __restrict__ xl1, float* __restrict__ xr1) {
  __shared__ float sWl[IN_F * F1], sWr[IN_F * F1], sbl[F1], sbr[F1], sx[2 * IN_F];
  int tid = threadIdx.x;                            // 256 threads, 2 nodes/block
  for (int i = tid; i < IN_F * F1; i += 256) { sWl[i] = Wl[i]; sWr[i] = Wr[i]; }
  if (tid < F1) { sbl[tid] = bl[tid]; sbr[tid] = br[tid]; }
  int nodeBase = blockIdx.x * 2;
  if (tid < 2 * IN_F) sx[tid] = x[nodeBase * IN_F + tid];
  __syncthreads();
  int local = tid >> 7, c = tid & 127;
  int node  = nodeBase + local;
  const float* xv = &sx[local * IN_F];
  float al = sbl[c], ar = sbr[c];
#pragma unroll
  for (int k = 0; k < IN_F; k++) {
    float xk = xv[k];
    al += xk * sWl[k * F1 + c];
    ar += xk * sWr[k * F1 + c];
  }
  xl1[node * F1 + c] = al;
  xr1[node * F1 + c] = ar;
}

// ------------------------------------------------- layer-1 edge pipeline
// pass 1: logits + segment-max.  One wave (32 lanes) per edge, lane = channel.
__global__ void edge1_logits_kernel(const int* __restrict__ ei,
                                    const float* __restrict__ eattr,
                                    const float* __restrict__ We,
                                    const float* __restrict__ att,
                                    const float* __restrict__ xl1,
                                    const float* __restrict__ xr1,
                                    float* __restrict__ alpha1,
                                    float* __restrict__ amax1) {
  __shared__ float sWe[ED_F * F1], satt[F1];
  int tid = threadIdx.x;
  for (int i = tid; i < ED_F * F1; i += 256) sWe[i] = We[i];
  if (tid < F1) satt[tid] = att[tid];
  __syncthreads();
  int e = blockIdx.x * 8 + (tid >> 5);
  int lane = tid & 31;
  int src = ei[e], dst = ei[EE + e];
  float ea0 = eattr[e * 4 + 0], ea1 = eattr[e * 4 + 1];
  float ea2 = eattr[e * 4 + 2], ea3 = eattr[e * 4 + 3];
  const float* xls = xl1 + (size_t)src * F1;
  const float* xrd = xr1 + (size_t)dst * F1;
#pragma unroll
  for (int h = 0; h < NHEADS; h++) {
    int f = h * HIDC + lane;
    float ef = ea0 * sWe[f] + ea1 * sWe[F1 + f] + ea2 * sWe[2 * F1 + f] + ea3 * sWe[3 * F1 + f];
    float z = xls[f] + xrd[f] + ef;
    z = z > 0.f ? z : 0.2f * z;                       // leaky_relu(0.2)
    float a = warp_sum(z * satt[f]);                  // einsum with att[h]
    if (lane == h) {
      alpha1[e * NHEADS + h] = a;
      atomicMaxF(&amax1[dst * NHEADS + h], a);
    }
  }
}

// pass 2: exp(logit - max), segment-sum.
__global__ void edge1_softmax_kernel(const int* __restrict__ ei,
                                     float* __restrict__ alpha1,
                                     const float* __restrict__ amax1,
                                     float* __restrict__ asum1) {
  int i = blockIdx.x * 256 + threadIdx.x;             // EE*NHEADS exact
  int e = i >> 2, h = i & 3;
  int dst = ei[EE + e];
  float a = __expf(alpha1[i] - amax1[dst * NHEADS + h]);
  alpha1[i] = a;
  atomicAdd(&asum1[dst * NHEADS + h], a);
}

// pass 3: out[dst] += xl1[src] * alpha/asum   (scatter-add, wave per edge)
__global__ void edge1_aggregate_kernel(const int* __restrict__ ei,
                                       const float* __restrict__ alpha1,
                                       const float* __restrict__ asum1,
                                       const float* __restrict__ xl1,
                                       float* __restrict__ acc1) {
  int e = blockIdx.x * 8 + (threadIdx.x >> 5);
  int lane = threadIdx.x & 31;
  int src = ei[e], dst = ei[EE + e];
  const float* xls = xl1 + (size_t)src * F1;
  float* accd = acc1 + (size_t)dst * F1;
#pragma unroll
  for (int h = 0; h < NHEADS; h++) {
    float w = alpha1[e * NHEADS + h] / (asum1[dst * NHEADS + h] + 1e-16f);
    int f = h * HIDC + lane;
    atomicAdd(&accd[f], xls[f] * w);
  }
}

// LayerNorm + ELU over 128 channels, in place.  Wave per node.
__global__ void ln1_kernel(float* __restrict__ acc1, const float* __restrict__ bias,
                           const float* __restrict__ g, const float* __restrict__ b) {
  __shared__ float sb[F1], sg[F1], sbb[F1];
  int tid = threadIdx.x;
  if (tid < F1) { sb[tid] = bias[tid]; sg[tid] = g[tid]; sbb[tid] = b[tid]; }
  __syncthreads();
  int n = blockIdx.x * 8 + (tid >> 5);
  int lane = tid & 31;
  float* row = acc1 + (size_t)n * F1;
  float v[4], s = 0.f;
#pragma unroll
  for (int i = 0; i < 4; i++) { v[i] = row[i * 32 + lane] + sb[i * 32 + lane]; s += v[i]; }
  float mu = warp_sum(s) * (1.f / F1);
  float q = 0.f;
#pragma unroll
  for (int i = 0; i < 4; i++) { float d = v[i] - mu; q += d * d; }
  float rstd = rsqrtf(warp_sum(q) * (1.f / F1) + 1e-5f);
#pragma unroll
  for (int i = 0; i < 4; i++) {
    int f = i * 32 + lane;
    float o = (v[i] - mu) * rstd * sg[f] + sbb[f];
    row[f] = o > 0.f ? o : (__expf(o) - 1.f);         // ELU
  }
}

// ------------------------------------------------- layer-2 node transform
// xl2 = h1@Wl2+bl2 ; xr2 = h1@Wr2+br2   via V_WMMA_F32_16X16X4_F32.
// Block = 4 waves; each wave owns one 16x16 tile of the fused [*,64] output
// (cols 0-31 -> Wl2, 32-63 -> Wr2).  K=128 -> 32 WMMA steps.
// Epilogue is branchless: one pointer select per lane, then 8 plain stores.
__global__ void gemm2_wmma_kernel(const float* __restrict__ h1,
                                  const float* __restrict__ Wl, const float* __restrict__ bl,
                                  const float* __restrict__ Wr, const float* __restrict__ br,
                                  float* __restrict__ xl2, float* __restrict__ xr2) {
  __shared__ float sW[F1 * 64];
  __shared__ float sbias[64];
  int tid = threadIdx.x;                              // 128
  for (int i = tid; i < F1 * 64; i += 128) {
    int k = i >> 6, c = i & 63;
    sW[i] = (c < HIDC) ? Wl[k * HIDC + c] : Wr[k * HIDC + (c - HIDC)];
  }
  if (tid < 64) sbias[tid] = (tid < HIDC) ? bl[tid] : br[tid - HIDC];
  __syncthreads();

  int wave = tid >> 5, lane = tid & 31;
  int n0 = wave * 16;
  size_t rowBase = (size_t)blockIdx.x * 16;           // NN = 3125*16 exactly
  int m    = lane & 15;
  int koff = (lane >> 4) * 2;                         // A/B lane layout (ISA 7.12.2)
  int ncol = lane & 15;
  const float* arow = h1 + (rowBase + m) * F1;

  v8f c = {};
#pragma unroll 4
  for (int kk = 0; kk < F1; kk += 4) {
    v2f a, b;
    a.x = arow[kk + koff];
    a.y = arow[kk + koff + 1];
    b.x = sW[(kk + koff) * 64 + n0 + ncol];
    b.y = sW[(kk + koff + 1) * 64 + n0 + ncol];
    c = __builtin_amdgcn_wmma_f32_16x16x4_f32(false, a, false, b, (short)0, c,
                                              false, false);
  }

  int rbase = (lane >> 4) * 8;                        // C/D: lanes 16-31 hold M=8..15
  int outc = n0 + ncol;
  float bia = sbias[outc];
  // single 64-bit pointer select instead of per-store divergent branches
  float* outp = (outc < HIDC) ? (xl2 + outc) : (xr2 + (outc - HIDC));
  outp += (rowBase + rbase) * HIDC;
#pragma unroll
  for (int i = 0; i < 8; i++) {
    outp[(size_t)i * HIDC] = c[i] + bia;
  }
}

// ------------------------------------------------- layer-2 edge pipeline
__global__ void edge2_logits_kernel(const int* __restrict__ ei,
                                    const float* __restrict__ eattr,
                                    const float* __restrict__ We2,
                                    const float* __restrict__ att2,
                                    const float* __restrict__ xl2,
                                    const float* __restrict__ xr2,
                                    float* __restrict__ alpha2,
                                    float* __restrict__ amax2) {
  __shared__ float sWe[ED_F * HIDC], satt[HIDC];
  int tid = threadIdx.x;
  if (tid < ED_F * HIDC) sWe[tid] = We2[tid];
  if (tid < HIDC) satt[tid] = att2[tid];
  __syncthreads();
  int e = blockIdx.x * 8 + (tid >> 5);
  int lane = tid & 31;
  int src = ei[e], dst = ei[EE + e];
  float ea0 = eattr[e * 4 + 0], ea1 = eattr[e * 4 + 1];
  float ea2 = eattr[e * 4 + 2], ea3 = eattr[e * 4 + 3];
  float ef = ea0 * sWe[lane] + ea1 * sWe[32 + lane] + ea2 * sWe[64 + lane] + ea3 * sWe[96 + lane];
  float z = xl2[(size_t)src * HIDC + lane] + xr2[(size_t)dst * HIDC + lane] + ef;
  z = z > 0.f ? z : 0.2f * z;
  float a = warp_sum(z * satt[lane]);
  if (lane == 0) {
    alpha2[e] = a;
    atomicMaxF(&amax2[dst], a);
  }
}

__global__ void edge2_softmax_kernel(const int* __restrict__ ei,
                                     float* __restrict__ alpha2,
                                     const float* __restrict__ amax2,
                                     float* __restrict__ asum2) {
  int e = blockIdx.x * 256 + threadIdx.x;             // EE exact
  int dst = ei[EE + e];
  float a = __expf(alpha2[e] - amax2[dst]);
  alpha2[e] = a;
  atomicAdd(&asum2[dst], a);
}

__global__ void edge2_aggregate_kernel(const int* __restrict__ ei,
                                       const float* __restrict__ alpha2,
                                       const float* __restrict__ asum2,
                                       const float* __restrict__ xl2,
                                       float* __restrict__ acc2) {
  int e = blockIdx.x * 8 + (threadIdx.x >> 5);
  int lane = threadIdx.x & 31;
  int src = ei[e], dst = ei[EE + e];
  float w = alpha2[e] / (asum2[dst] + 1e-16f);
  atomicAdd(&acc2[(size_t)dst * HIDC + lane],
            xl2[(size_t)src * HIDC + lane] * w);
}

// LN+ELU over 32 channels -> d_out rows, fused with mean/max pooling partials.
__global__ void ln2_pool_kernel(const float* __restrict__ acc2,
                                const float* __restrict__ bias2,
                                const float* __restrict__ g2,
                                const float* __restrict__ b2,
                                float* __restrict__ out_h,
                                float* __restrict__ gsum, float* __restrict__ gmax) {
  __shared__ float sred[8][32], sredm[8][32];
  int tid = threadIdx.x, wave = tid >> 5, lane = tid & 31;
  float bia = bias2[lane], gg = g2[lane], bb = b2[lane];
  float psum = 0.f, pmax = -__builtin_inff();
  int stride = gridDim.x * 8;
  for (int n = blockIdx.x * 8 + wave; n < NN; n += stride) {
    float v  = acc2[(size_t)n * HIDC + lane] + bia;
    float mu = warp_sum(v) * (1.f / HIDC);
    float d  = v - mu;
    float rstd = rsqrtf(warp_sum(d * d) * (1.f / HIDC) + 1e-5f);
    float o = d * rstd * gg + bb;
    o = o > 0.f ? o : (__expf(o) - 1.f);
    out_h[(size_t)n * HIDC + lane] = o;
    psum += o;
    pmax = fmaxf(pmax, o);
  }
  sred[wave][lane] = psum;
  sredm[wave][lane] = pmax;
  __syncthreads();
  if (wave == 0) {
    float s = sred[0][lane], m = sredm[0][lane];
#pragma unroll
    for (int w2 = 1; w2 < 8; w2++) { s += sred[w2][lane]; m = fmaxf(m, sredm[w2][lane]); }
    atomicAdd(&gsum[lane], s);
    atomicMaxF(&gmax[lane], m);
  }
}

__global__ void finalize_pool_kernel(const float* __restrict__ gsum,
                                     const float* __restrict__ gmax,
                                     float* __restrict__ gout) {
  int t = threadIdx.x;                                // 64 threads
  gout[t] = (t < HIDC) ? gsum[t] * (1.f / NN) : gmax[t - HIDC];
}

// ---------------------------------------------------------------- launch
extern "C" void kernel_launch(void* const* d_in, const int* in_sizes, int n_in,
                              void* d_out, int out_size, void* d_ws, size_t ws_size,
                              hipStream_t stream) {
  const float* x     = (const float*)d_in[0];
  const int*   ei    = (const int*)  d_in[1];   // [2,E]
  const float* eattr = (const float*)d_in[2];
  const float* Wl1   = (const float*)d_in[3];
  const float* bl1   = (const float*)d_in[4];
  const float* Wr1   = (const float*)d_in[5];
  const float* br1   = (const float*)d_in[6];
  const float* We1   = (const float*)d_in[7];
  const float* att1  = (const float*)d_in[8];
  const float* bias1 = (const float*)d_in[9];
  const float* ln1g  = (const float*)d_in[10];
  const float* ln1b  = (const float*)d_in[11];
  const float* Wl2   = (const float*)d_in[12];
  const float* bl2   = (const float*)d_in[13];
  const float* Wr2   = (const float*)d_in[14];
  const float* br2   = (const float*)d_in[15];
  const float* We2   = (const float*)d_in[16];
  const float* att2  = (const float*)d_in[17];
  const float* bias2 = (const float*)d_in[18];
  const float* ln2g  = (const float*)d_in[19];
  const float* ln2b  = (const float*)d_in[20];

  float* out_h = (float*)d_out;                 // [N,32]
  float* out_g = out_h + (size_t)NN * HIDC;     // [64]

  // workspace layout (floats); layer-2 tensors reuse the xl1 region.
  float* ws = (float*)d_ws;
  float* xl1    = ws;
  float* xr1    = ws + (size_t)NN * F1;
  float* acc1   = ws + 2 * (size_t)NN * F1;     // becomes h1 in place after LN1
  float* alpha  = ws + 3 * (size_t)NN * F1;     // E*4 (layer1) / E (layer2)
  float* amax1  = alpha + (size_t)EE * NHEADS;
  float* asum1  = amax1 + (size_t)NN * NHEADS;
  float* amax2  = asum1 + (size_t)NN * NHEADS;
  float* asum2  = amax2 + NN;
  float* gsum   = asum2 + NN;
  float* gmax   = gsum + HIDC;
  float* xl2    = ws;                            // reuse (layer1 xl1 dead)
  float* xr2    = ws + (size_t)NN * HIDC;
  float* acc2   = ws + 2 * (size_t)NN * HIDC;

  // ---- layer 1 ----
  init1_kernel<<<(NN * F1 + 255) / 256, 256, 0, stream>>>(acc1, amax1, asum1);
  gemm1_kernel<<<NN / 2, 256, 0, stream>>>(x, Wl1, bl1, Wr1, br1, xl1, xr1);
  edge1_logits_kernel<<<EE / 8, 256, 0, stream>>>(ei, eattr, We1, att1, xl1, xr1,
                                                  alpha, amax1);
  edge1_softmax_kernel<<<EE * NHEADS / 256, 256, 0, stream>>>(ei, alpha, amax1, asum1);
  edge1_aggregate_kernel<<<EE / 8, 256, 0, stream>>>(ei, alpha, asum1, xl1, acc1);
  ln1_kernel<<<NN / 8, 256, 0, stream>>>(acc1, bias1, ln1g, ln1b);

  // ---- layer 2 ----
  init2_kernel<<<(NN * HIDC + 255) / 256, 256, 0, stream>>>(acc2, amax2, asum2,
                                                            gsum, gmax);
  gemm2_wmma_kernel<<<NN / 16, 128, 0, stream>>>(acc1, Wl2, bl2, Wr2, br2, xl2, xr2);
  edge2_logits_kernel<<<EE / 8, 256, 0, stream>>>(ei, eattr, We2, att2, xl2, xr2,
                                                  alpha, amax2);
  edge2_softmax_kernel<<<EE / 256, 256, 0, stream>>>(ei, alpha, amax2, asum2);
  edge2_aggregate_kernel<<<EE / 8, 256, 0, stream>>>(ei, alpha, asum2, xl2, acc2);
  ln2_pool_kernel<<<512, 256, 0, stream>>>(acc2, bias2, ln2g, ln2b, out_h, gsum, gmax);
  finalize_pool_kernel<<<1, 64, 0, stream>>>(gsum, gmax, out_g);
}